// LeakyAvg_11725260718268
// MI455X (gfx1250) — compile-verified
//
#include <hip/hip_runtime.h>

typedef float v2f __attribute__((ext_vector_type(2)));
typedef float v8f __attribute__((ext_vector_type(8)));

// LeakyAvg: out[t] = k[t] + a*out[t-1], a = exp(-|beta_h|*10)
// k: (B=8, H=16, T=4096, D=64) f32.  Memory-bound: 256 MiB total -> ~11.5us floor.
//
// 16-step scan tile == lower-triangular 16x16 f32 matmul (V_WMMA_F32_16X16X4_F32 x4),
// inter-tile carry injected as a 5th rank-1 WMMA; carry broadcast via ds_bpermute.
// Data movement: per-wave 4-deep LDS ring fed by global_load_async_to_lds_b128
// (ASYNCcnt pipeline, 16B/lane), plus far global_prefetch into L2 (~32 tiles ahead)
// so demand async loads hit L2 instead of raw HBM.

#define ASYNC_LOAD_B128(ldsoff, voff, sbase)                          \
    asm volatile("global_load_async_to_lds_b128 %0, %1, %2 offset:0"  \
                 :: "v"(ldsoff), "v"(voff), "s"(sbase) : "memory")

#if __has_builtin(__builtin_amdgcn_s_wait_asynccnt)
#define WAIT_ASYNCCNT(n) __builtin_amdgcn_s_wait_asynccnt(n)
#else
#define WAIT_ASYNCCNT(n) asm volatile("s_wait_asynccnt %0" :: "n"(n) : "memory")
#endif
#define WAIT_DSCNT0() asm volatile("s_wait_dscnt 0x0" ::: "memory")

__global__ __launch_bounds__(128) void leaky_scan_wmma(
    const float* __restrict__ k,
    const float* __restrict__ beta,
    float* __restrict__ out)
{
    constexpr int H = 16, T = 4096, D = 64;
    constexpr int NTILES = T / 16;      // 256
    constexpr int NBUF   = 4;           // per-wave LDS ring depth

    const int bh   = blockIdx.x;        // 0..127  (b*H + h)
    const int h    = bh & (H - 1);
    const int lane = threadIdx.x & 31;
    const int wave = threadIdx.x >> 5;  // 0..3
    const int d0   = wave * 16;
    const int col  = lane & 15;
    const int hi   = lane >> 4;         // 0: lanes 0-15, 1: lanes 16-31

    // 4 waves x 4 bufs x (16 rows x 16 floats)
    __shared__ float smem[4 * NBUF * 256];
    const unsigned lds_wave =
        (unsigned)(uintptr_t)(&smem[0]) + (unsigned)wave * (NBUF * 1024);

    // a = exp(-|beta|*10); a^n computed exactly as exp(n * -|beta|*10)
    const float nb = -fabsf(beta[h]) * 10.0f;

    // A-operands for M (16x16 lower-triangular) split into 4 K=4 chunks.
    // f32 16x4 A layout (wave32): lane holds M=col; V0: K=4c+2*hi, V1: K=4c+2*hi+1.
    v2f aM[4];
    #pragma unroll
    for (int c = 0; c < 4; ++c) {
        const int j0 = 4 * c + 2 * hi;
        aM[c].x = (col >= j0)     ? __expf(nb * (float)(col - j0))     : 0.0f;
        aM[c].y = (col >= j0 + 1) ? __expf(nb * (float)(col - j0 - 1)) : 0.0f;
    }
    // Carry injector: only K-column 0 nonzero, value a^(i+1).
    v2f aP;
    aP.x = (hi == 0) ? __expf(nb * (float)(col + 1)) : 0.0f;
    aP.y = 0.0f;

    const float* __restrict__ kbh = k   + (size_t)bh * (T * D);
    float*       __restrict__ obh = out + (size_t)bh * (T * D);

    // --- stage one 16t x 16d wave-slice into LDS ring buffer `buf` (2x b128/lane)
    auto stage = [&](int tile, int buf) {
        const unsigned row = (unsigned)(lane >> 2);        // 0..7
        const unsigned seg = (unsigned)(lane & 3) * 16u;   // 16B seg in 64B row
        const unsigned voff =
            (unsigned)(((tile * 16 + (int)row) * D + d0) * 4) + seg;
        const unsigned loff = lds_wave + (unsigned)buf * 1024u + row * 64u + seg;
        ASYNC_LOAD_B128(loff, voff, kbh);                      // rows 0-7
        ASYNC_LOAD_B128(loff + 512u, voff + 8u * D * 4u, kbh); // rows 8-15
    };

    float carry = 0.0f;  // column d0+col carry, valid in lanes 0-15

    // --- consume one tile from LDS ring buffer `buf`
    auto consume = [&](int tile, int buf) {
        const float* wbuf = &smem[wave * NBUF * 256 + buf * 256];
        v2f bB[4];
        #pragma unroll
        for (int c = 0; c < 4; ++c) {
            const int r0 = 4 * c + 2 * hi;
            bB[c].x = wbuf[r0 * 16 + col];
            bB[c].y = wbuf[(r0 + 1) * 16 + col];
        }
        v8f acc = {};
        #pragma unroll
        for (int c = 0; c < 4; ++c) {
            acc = __builtin_amdgcn_wmma_f32_16x16x4_f32(
                false, aM[c], false, bB[c], (short)0, acc, false, false);
        }
        v2f bC;
        bC.x = (hi == 0) ? carry : 0.0f;
        bC.y = 0.0f;
        acc = __builtin_amdgcn_wmma_f32_16x16x4_f32(
            false, aP, false, bC, (short)0, acc, false, false);

        const int t0 = tile * 16;
        #pragma unroll
        for (int v = 0; v < 8; ++v) {
            obh[(t0 + v + 8 * hi) * D + d0 + col] = acc[v];
        }
        carry = __shfl(acc[7], col + 16, 32);
    };

    // Prime the ring.
    #pragma unroll
    for (int b = 0; b < NBUF; ++b) stage(b, b);

    int tile = 0;
    for (; tile < NTILES - NBUF; ++tile) {
        // Far prefetch into L2 (~32 tiles ahead); demand async loads then hit L2.
        if (tile < NTILES - 32) {
            __builtin_prefetch(&kbh[(tile * 16 + 512 + col) * D + d0], 0, 0);
        }
        // Oldest stage (2 async ops) retired once <=6 of the <=8 remain.
        WAIT_ASYNCCNT(6);
        consume(tile, tile & (NBUF - 1));
        // Ensure ds reads of this buffer are done before async overwrite.
        WAIT_DSCNT0();
        stage(tile + NBUF, tile & (NBUF - 1));
    }
    // Tail: everything already staged.
    WAIT_ASYNCCNT(0);
    for (; tile < NTILES; ++tile) {
        consume(tile, tile & (NBUF - 1));
    }
}

extern "C" void kernel_launch(void* const* d_in, const int* in_sizes, int n_in,
                              void* d_out, int out_size, void* d_ws, size_t ws_size,
                              hipStream_t stream) {
    const float* k    = (const float*)d_in[0];
    const float* beta = (const float*)d_in[1];
    float* out        = (float*)d_out;

    dim3 grid(128);   // B*H = 8*16
    dim3 block(128);  // 4 waves of 32
    hipLaunchKernelGGL(leaky_scan_wmma, grid, block, 0, stream, k, beta, out);
}